// Net_38147899523750
// MI455X (gfx1250) — compile-verified
//
#include <hip/hip_runtime.h>
#include <hip/hip_bf16.h>
#include <math.h>

// ---------------------------------------------------------------------------
// GraphConv x2 + linear head + log_softmax for MI455X (gfx1250, wave32).
// Dense GEMMs use V_WMMA_F32_16X16X4_F32 (fp32; workload is scatter-bandwidth
// bound so fp32 matrix ops are free). Weights are pre-packed once so each
// lane's B fragment (two K-adjacent values of one column) is a single
// contiguous 8-byte load (global_load_b64) in the WMMA inner loop.
// ---------------------------------------------------------------------------

typedef __attribute__((ext_vector_type(2))) float v2f;
typedef __attribute__((ext_vector_type(8))) float v8f;

#define N_NODES   50000
#define ROW_TILES (N_NODES / 16)   // 50000 = 16 * 3125 exactly

// ---------------------------------------------------------------------------
// Weight repack: B[K, n] row-major -> Bp[kp][n] of v2f, where
//   Bp[kp*n + col] = (B[2kp][col], B[2kp+1][col]).
// One-time, ~868 KB total across all five weight matrices.
// ---------------------------------------------------------------------------
__global__ void pack_b_kernel(const float* __restrict__ B,
                              float* __restrict__ Bp, int K, int n) {
  const int i = blockIdx.x * blockDim.x + threadIdx.x;  // over (K/2)*n
  const int total = (K / 2) * n;
  if (i >= total) return;
  const int kp = i / n;
  const int col = i - kp * n;
  Bp[(size_t)i * 2 + 0] = B[(size_t)(2 * kp) * n + col];
  Bp[(size_t)i * 2 + 1] = B[(size_t)(2 * kp + 1) * n + col];
}

// ---------------------------------------------------------------------------
// WMMA accumulate: acc[t] += A[m0:m0+16, 0:K] @ B[0:K, n0+16t : n0+16t+16]
// A row-major [*, lda]; Bp is the packed weight (v2f rows of length nOut).
// Column reads clamped to nmax-1 so padded tiles never read OOB (results
// discarded by the guarded store).
//
// f32 16x16x4 fragment layout (ISA 7.12.2):
//   A: lane h=lane>>4, m=lane&15; v0=A[m][2h], v1=A[m][2h+1]
//   B: symmetric over columns:    v0=B[2h][n], v1=B[2h+1][n]
//   C/D: 8 VGPRs; vr = D[r + 8h][lane&15]
// ---------------------------------------------------------------------------
template <int CT>
__device__ inline void wmma_acc_f32(const float* __restrict__ A, int lda,
                                    const v2f* __restrict__ Bp, int nOut,
                                    int nmax, int K, int m0, int n0, v8f* acc) {
  const int lane = threadIdx.x & 31;
  const int half = lane >> 4;
  const float* __restrict__ arow = A + (size_t)(m0 + (lane & 15)) * lda;

  int ncol[CT];
#pragma unroll
  for (int t = 0; t < CT; ++t) {
    int n = n0 + (lane & 15) + 16 * t;
    ncol[t] = (n < nmax) ? n : (nmax - 1);  // clamp keeps Bp reads in-bounds
  }

  for (int k = 0; k < K; k += 4) {
    const int ka = k + 2 * half;      // even
    const int kp = ka >> 1;           // packed row index
    v2f a;
    a.x = arow[ka];
    a.y = arow[ka + 1];               // contiguous -> b64 load
    const v2f* __restrict__ brow = Bp + (size_t)kp * nOut;
#pragma unroll
    for (int t = 0; t < CT; ++t) {
      const v2f b = brow[ncol[t]];    // contiguous pair -> b64 load
      acc[t] = __builtin_amdgcn_wmma_f32_16x16x4_f32(
          /*neg_a=*/false, a, /*neg_b=*/false, b,
          /*c_mod=*/(short)0, acc[t], /*reuse_a=*/false, /*reuse_b=*/false);
    }
  }
}

// ---------------------------------------------------------------------------
// C = act(A1@B1 + A2@B2 + bias), one wave per 16 x (16*CT) output strip.
// B1p/B2p are packed weights.
// ---------------------------------------------------------------------------
template <int CT>
__global__ void gemm2term_kernel(const float* __restrict__ A1, int K1,
                                 const v2f* __restrict__ B1p,
                                 const float* __restrict__ A2, int K2,
                                 const v2f* __restrict__ B2p,
                                 const float* __restrict__ bias,
                                 float* __restrict__ C, int nOut, int ldc,
                                 int nmax, int relu) {
  const int wave = threadIdx.x >> 5;
  const int rt = blockIdx.x * (blockDim.x >> 5) + wave;
  if (rt >= ROW_TILES) return;  // wave-uniform: EXEC stays all-ones for WMMA
  const int m0 = rt * 16;
  const int n0 = blockIdx.y * (16 * CT);

  v8f acc[CT];
#pragma unroll
  for (int t = 0; t < CT; ++t) acc[t] = (v8f){0, 0, 0, 0, 0, 0, 0, 0};

  wmma_acc_f32<CT>(A1, K1, B1p, nOut, nmax, K1, m0, n0, acc);
  wmma_acc_f32<CT>(A2, K2, B2p, nOut, nmax, K2, m0, n0, acc);

  const int lane = threadIdx.x & 31;
  const int half = lane >> 4;
#pragma unroll
  for (int t = 0; t < CT; ++t) {
    const int n = n0 + (lane & 15) + 16 * t;
    if (n >= nmax) continue;           // padded columns: discard
    const float bv = bias[n];
#pragma unroll
    for (int r = 0; r < 8; ++r) {
      const int m = m0 + r + 8 * half;
      float v = acc[t][r] + bv;
      if (relu) v = fmaxf(v, 0.0f);
      C[(size_t)m * ldc + n] = v;
    }
  }
}

// ---------------------------------------------------------------------------
// Scatter-add: one wave32 per edge; float4 gather of x[src], fp32 atomics
// into agg[dst]. F = 128 or 256 (multiple of 128).
// ---------------------------------------------------------------------------
template <int F>
__global__ void scatter_add_kernel(const float* __restrict__ x,
                                   const int* __restrict__ ei,
                                   const float* __restrict__ ew,
                                   float* __restrict__ agg, int E) {
  const int e = (int)((blockIdx.x * (size_t)blockDim.x + threadIdx.x) >> 5);
  if (e >= E) return;
  const int lane = threadIdx.x & 31;
  const int s = ei[e];       // edge_index[0, e]
  const int d = ei[E + e];   // edge_index[1, e]
  const float wt = ew[e];
  const float4* __restrict__ xs = (const float4*)(x + (size_t)s * F);
  float* __restrict__ ag = agg + (size_t)d * F;
#pragma unroll
  for (int i = 0; i < F / 128; ++i) {
    const float4 v = xs[lane + 32 * i];
    const int base = (lane + 32 * i) * 4;
    atomicAdd(ag + base + 0, v.x * wt);
    atomicAdd(ag + base + 1, v.y * wt);
    atomicAdd(ag + base + 2, v.z * wt);
    atomicAdd(ag + base + 3, v.w * wt);
  }
}

// ---------------------------------------------------------------------------
// In-place row-wise log_softmax over 40 logits; one wave32 per row.
// ---------------------------------------------------------------------------
__global__ void log_softmax_kernel(float* __restrict__ out, int n) {
  const int row = (int)((blockIdx.x * (size_t)blockDim.x + threadIdx.x) >> 5);
  if (row >= n) return;
  const int lane = threadIdx.x & 31;
  float* __restrict__ p = out + (size_t)row * 40;
  const float v0 = p[lane];                          // lane < 32 < 40: valid
  const float v1 = (lane < 8) ? p[32 + lane] : -INFINITY;
  float m = fmaxf(v0, v1);
#pragma unroll
  for (int off = 16; off > 0; off >>= 1) m = fmaxf(m, __shfl_xor(m, off, 32));
  float s = expf(v0 - m) + ((lane < 8) ? expf(v1 - m) : 0.0f);
#pragma unroll
  for (int off = 16; off > 0; off >>= 1) s += __shfl_xor(s, off, 32);
  const float lse = m + logf(s);
  p[lane] = v0 - lse;
  if (lane < 8) p[32 + lane] = v1 - lse;
}

// ---------------------------------------------------------------------------
// Host-side orchestration. Inputs (setup_inputs order):
//  0 x[N,128] 1 edge_index[2,E] 2 edge_weight[E] 3 w1_rel[128,256] 4 b1[256]
//  5 w1_root[128,256] 6 w2_rel[256,256] 7 b2[256] 8 w2_root[256,256]
//  9 w_lin[512,40] 10 b_lin[40]
// ---------------------------------------------------------------------------
extern "C" void kernel_launch(void* const* d_in, const int* in_sizes, int n_in,
                              void* d_out, int out_size, void* d_ws,
                              size_t ws_size, hipStream_t stream) {
  const float* x       = (const float*)d_in[0];
  const int*   ei      = (const int*)d_in[1];
  const float* ew      = (const float*)d_in[2];
  const float* w1_rel  = (const float*)d_in[3];
  const float* b1      = (const float*)d_in[4];
  const float* w1_root = (const float*)d_in[5];
  const float* w2_rel  = (const float*)d_in[6];
  const float* b2      = (const float*)d_in[7];
  const float* w2_root = (const float*)d_in[8];
  const float* w_lin   = (const float*)d_in[9];
  const float* b_lin   = (const float*)d_in[10];
  float* out = (float*)d_out;

  const int N = N_NODES;
  const int E = in_sizes[2];  // edge_weight element count

  // Workspace layout (~180 MB; fits the 192 MB L2):
  float* agg1 = (float*)d_ws;                    // [N,128]
  float* x1   = agg1 + (size_t)N * 128;          // [N,256]
  float* agg2 = x1   + (size_t)N * 256;          // [N,256]
  float* x2   = agg2 + (size_t)N * 256;          // [N,256]
  // Packed weights (same element counts as originals, ~868 KB total):
  float* p1_rel  = x2      + (size_t)N * 256;    // 128*256
  float* p1_root = p1_rel  + 128 * 256;          // 128*256
  float* p2_rel  = p1_root + 128 * 256;          // 256*256
  float* p2_root = p2_rel  + 256 * 256;          // 256*256
  float* p_lin_a = p2_root + 256 * 256;          // 256*40 (w_lin rows 0:256)
  float* p_lin_b = p_lin_a + 256 * 40;           // 256*40 (w_lin rows 256:512)

  // ---- One-time weight repack (K/2 * n threads each)
  {
    const int T = 256;
    pack_b_kernel<<<(64 * 256 + T - 1) / T, T, 0, stream>>>(w1_rel, p1_rel, 128, 256);
    pack_b_kernel<<<(64 * 256 + T - 1) / T, T, 0, stream>>>(w1_root, p1_root, 128, 256);
    pack_b_kernel<<<(128 * 256 + T - 1) / T, T, 0, stream>>>(w2_rel, p2_rel, 256, 256);
    pack_b_kernel<<<(128 * 256 + T - 1) / T, T, 0, stream>>>(w2_root, p2_root, 256, 256);
    pack_b_kernel<<<(128 * 40 + T - 1) / T, T, 0, stream>>>(w_lin, p_lin_a, 256, 40);
    pack_b_kernel<<<(128 * 40 + T - 1) / T, T, 0, stream>>>(w_lin + (size_t)256 * 40,
                                                            p_lin_b, 256, 40);
  }

  const int sc_threads = 256;
  const int sc_blocks = (int)(((size_t)E * 32 + sc_threads - 1) / sc_threads);
  const dim3 gemm_grid((ROW_TILES + 7) / 8, 256 / 64);  // 8 waves/block, 4 col grps
  const dim3 head_grid((ROW_TILES + 7) / 8, 1);

  // ---- Layer 1: agg1 = scatter_add(x*w); x1 = relu(agg1@w1_rel + x@w1_root + b1)
  hipMemsetAsync(agg1, 0, (size_t)N * 128 * sizeof(float), stream);
  scatter_add_kernel<128><<<sc_blocks, sc_threads, 0, stream>>>(x, ei, ew, agg1, E);
  gemm2term_kernel<4><<<gemm_grid, 256, 0, stream>>>(
      agg1, 128, (const v2f*)p1_rel, x, 128, (const v2f*)p1_root, b1, x1,
      /*nOut=*/256, /*ldc=*/256, /*nmax=*/256, /*relu=*/1);

  // ---- Layer 2: agg2 = scatter_add(x1*w); x2 = relu(agg2@w2_rel + x1@w2_root + b2)
  hipMemsetAsync(agg2, 0, (size_t)N * 256 * sizeof(float), stream);
  scatter_add_kernel<256><<<sc_blocks, sc_threads, 0, stream>>>(x1, ei, ew, agg2, E);
  gemm2term_kernel<4><<<gemm_grid, 256, 0, stream>>>(
      agg2, 256, (const v2f*)p2_rel, x1, 256, (const v2f*)p2_root, b2, x2,
      /*nOut=*/256, /*ldc=*/256, /*nmax=*/256, /*relu=*/1);

  // ---- Head: logits = x1@w_lin[0:256] + x2@w_lin[256:512] + b_lin -> d_out
  gemm2term_kernel<3><<<head_grid, 256, 0, stream>>>(
      x1, 256, (const v2f*)p_lin_a, x2, 256, (const v2f*)p_lin_b, b_lin, out,
      /*nOut=*/40, /*ldc=*/40, /*nmax=*/40, /*relu=*/0);

  // ---- log_softmax in place on d_out
  const int ls_blocks = (int)(((size_t)N * 32 + 255) / 256);
  log_softmax_kernel<<<ls_blocks, 256, 0, stream>>>(out, N);
}